// GHMC_80195629351056
// MI455X (gfx1250) — compile-verified
//
#include <hip/hip_runtime.h>
#include <stdint.h>

#define TPB    256
#define WAVES  (TPB / 32)
#define NBINS  10
#define BLOCKS 2048
#define DEPTH  4          // async pipeline depth (buffers per wave)

// ---------------------------------------------------------------------------
// CDNA5 async global -> LDS copy (128b per lane), tracked by ASYNCcnt.
// gfx1250 MC form: global_load_async_to_lds_b128 vdst(lds addr), vaddr, off
// ---------------------------------------------------------------------------
__device__ __forceinline__ void async_copy_b128(uint32_t lds_off, unsigned long long gaddr) {
    asm volatile("global_load_async_to_lds_b128 %0, %1, off"
                 :: "v"(lds_off), "v"(gaddr)
                 : "memory");
}
__device__ __forceinline__ void wait_async_3() { asm volatile("s_wait_asynccnt 3" ::: "memory"); }
__device__ __forceinline__ void wait_async_2() { asm volatile("s_wait_asynccnt 2" ::: "memory"); }
__device__ __forceinline__ void wait_async_1() { asm volatile("s_wait_asynccnt 1" ::: "memory"); }
__device__ __forceinline__ void wait_async_0() { asm volatile("s_wait_asynccnt 0" ::: "memory"); }

// ---------------------------------------------------------------------------
// Per-element GHM-C work: g = |sigmoid(x) - t|; bin = clip(floor(10g),0,9);
// bce = max(x,0) - x*t + log1p(exp(-|x|)).
// 3 trans ops / element: v_exp_f32, v_rcp_f32, v_log_f32.
// ---------------------------------------------------------------------------
__device__ __forceinline__ void proc1(float x, float t,
                                      unsigned* __restrict__ my_cnt,
                                      float* __restrict__ my_sum) {
    const float LOG2E = 1.4426950408889634f;
    const float LN2   = 0.6931471805599453f;
    float ax = fabsf(x);
    float em = __builtin_amdgcn_exp2f(-ax * LOG2E);   // e^{-|x|} in (0,1]
    float r  = __builtin_amdgcn_rcpf(1.0f + em);      // sigmoid(|x|)
    float sg = (x >= 0.0f) ? r : em * r;              // sigmoid(x); 1-r == em*r
    float g  = fabsf(sg - t);
    int bin  = (int)(g * 10.0f);
    bin = bin > (NBINS - 1) ? (NBINS - 1) : bin;      // g >= 0 always
    float sp  = __builtin_amdgcn_logf(1.0f + em) * LN2;  // log1p(e^{-|x|})
    float bce = fmaxf(x, 0.0f) - x * t + sp;
    // Per-thread private LDS sub-histogram: single-instruction ds atomics,
    // bank-conflict-free (stride 10 dwords across lanes of a wave32).
    atomicAdd(&my_cnt[bin], 1u);
    atomicAdd(&my_sum[bin], bce);
}

__device__ __forceinline__ void proc4(float4 v, int tg, int col0,
                                      unsigned* __restrict__ c, float* __restrict__ s) {
    proc1(v.x, (tg == col0 + 0) ? 1.0f : 0.0f, c, s);
    proc1(v.y, (tg == col0 + 1) ? 1.0f : 0.0f, c, s);
    proc1(v.z, (tg == col0 + 2) ? 1.0f : 0.0f, c, s);
    proc1(v.w, (tg == col0 + 3) ? 1.0f : 0.0f, c, s);
}

// ---------------------------------------------------------------------------
// Pass 1: stream pred through LDS with a DEPTH-deep async pipeline,
// accumulate per-bin {count, sum(bce)} in per-thread LDS histograms,
// tree-reduce per block, one global atomic per bin per block.
// ---------------------------------------------------------------------------
__global__ __launch_bounds__(TPB) void ghmc_pass1(
    const float* __restrict__ pred, const int* __restrict__ target,
    unsigned* __restrict__ g_cnt, float* __restrict__ g_sum,
    long n4, long S, int shiftC, int maskC)
{
    __shared__ float4   stage[WAVES * DEPTH * 32];  // 16 KB: DEPTH x 512B per wave
    __shared__ unsigned s_cnt[TPB * NBINS];         // 10 KB
    __shared__ float    s_sum[TPB * NBINS];         // 10 KB

    const int tid  = threadIdx.x;
    const int lane = tid & 31;
    const int wave = tid >> 5;

    unsigned* my_cnt = &s_cnt[tid * NBINS];
    float*    my_sum = &s_sum[tid * NBINS];
#pragma unroll
    for (int b = 0; b < NBINS; ++b) { my_cnt[b] = 0u; my_sum[b] = 0.0f; }

    const long gid        = (long)blockIdx.x * TPB + tid;
    const long lanesTotal = (long)gridDim.x * TPB;

    // LDS byte offset of this lane's slot in buffer 0 (low 32 bits of the
    // generic pointer to a __shared__ object are the LDS byte address).
    // Slot k is at +k*512 bytes.
    const uint32_t lds_base =
        (uint32_t)(uintptr_t)(&stage[(wave * DEPTH) * 32 + lane]);

    const float4* gp = (const float4*)pred;

    if (S > 0) {
        // Prologue: fill up to DEPTH-1 buffers.
#pragma unroll
        for (int p = 0; p < DEPTH - 1; ++p) {
            if (p < S)
                async_copy_b128(lds_base + (uint32_t)p * 512u,
                                (unsigned long long)(gp + gid + (long)p * lanesTotal));
        }
        for (long s = 0; s < S; ++s) {
            const int buf = (int)(s & (DEPTH - 1));
            if (s + (DEPTH - 1) < S) {
                const int nbuf = (int)((s + (DEPTH - 1)) & (DEPTH - 1));
                async_copy_b128(lds_base + (uint32_t)nbuf * 512u,
                                (unsigned long long)(gp + gid + (s + (DEPTH - 1)) * lanesTotal));
                wait_async_3();              // issued = s+DEPTH; <=3 left => stage s done
            } else {
                const long rem = S - 1 - s;  // stages still in flight after current
                if      (rem >= 2) wait_async_2();
                else if (rem == 1) wait_async_1();
                else               wait_async_0();
            }
            float4 v  = stage[(wave * DEPTH + buf) * 32 + lane];
            long   f  = gid + s * lanesTotal;
            long   e0 = f << 2;
            int row  = (int)(e0 >> shiftC);
            int col0 = (int)(e0 & (long)maskC);
            proc4(v, target[row], col0, my_cnt, my_sum);
        }
    }
    // Generic tail (n4 not divisible by lanes*S): plain vector loads.
    for (long f = lanesTotal * S + gid; f < n4; f += lanesTotal) {
        float4 v  = gp[f];
        long   e0 = f << 2;
        int row  = (int)(e0 >> shiftC);
        int col0 = (int)(e0 & (long)maskC);
        proc4(v, target[row], col0, my_cnt, my_sum);
    }

    __syncthreads();
    // Tree-reduce the 256 per-thread sub-histograms.
    for (int str = TPB / 2; str > 0; str >>= 1) {
        if (tid < str) {
#pragma unroll
            for (int b = 0; b < NBINS; ++b) {
                s_cnt[tid * NBINS + b] += s_cnt[(tid + str) * NBINS + b];
                s_sum[tid * NBINS + b] += s_sum[(tid + str) * NBINS + b];
            }
        }
        __syncthreads();
    }
    if (tid < NBINS) {
        unsigned c = s_cnt[tid];
        if (c) {
            atomicAdd(&g_cnt[tid], c);
            atomicAdd(&g_sum[tid], s_sum[tid]);
        }
    }
}

// ---------------------------------------------------------------------------
// Zero the global accumulators (d_ws is poisoned by the harness).
// ---------------------------------------------------------------------------
__global__ void ghmc_zero(unsigned* __restrict__ g_cnt, float* __restrict__ g_sum) {
    int t = threadIdx.x;
    if (t < NBINS) { g_cnt[t] = 0u; g_sum[t] = 0.0f; }
}

// ---------------------------------------------------------------------------
// Finalize: loss = (1/n) * sum_b (count_b>0 ? sum_b/count_b : 0),
// n = max(#nonempty, 1).  (tot cancels out of the reference formula.)
// ---------------------------------------------------------------------------
__global__ void ghmc_finalize(const unsigned* __restrict__ g_cnt,
                              const float* __restrict__ g_sum,
                              float* __restrict__ out) {
    int l = threadIdx.x;           // single wave32
    float contrib = 0.0f;
    int   ne      = 0;
    if (l < NBINS) {
        unsigned c = g_cnt[l];
        if (c > 0u) { contrib = g_sum[l] / (float)c; ne = 1; }
    }
#pragma unroll
    for (int off = 16; off > 0; off >>= 1) {
        contrib += __shfl_down(contrib, off, 32);
        ne      += __shfl_down(ne, off, 32);
    }
    if (l == 0) {
        int n = ne > 1 ? ne : 1;
        out[0] = contrib / (float)n;   // LOSS_WEIGHT == 1.0
    }
}

// ---------------------------------------------------------------------------
extern "C" void kernel_launch(void* const* d_in, const int* in_sizes, int n_in,
                              void* d_out, int out_size, void* d_ws, size_t ws_size,
                              hipStream_t stream) {
    const float* pred   = (const float*)d_in[0];
    const int*   target = (const int*)d_in[1];

    const long total = (long)in_sizes[0];          // N*C
    const int  N     = in_sizes[1];
    const int  C     = (int)(total / N);           // 1024 (power of two)
    const int  shiftC = 31 - __builtin_clz((unsigned)C);
    const int  maskC  = C - 1;

    unsigned* g_cnt = (unsigned*)d_ws;
    float*    g_sum = (float*)((char*)d_ws + 64);  // 64B-aligned second array

    const long n4    = total >> 2;                 // float4 count
    const long lanes = (long)BLOCKS * TPB;
    const long S     = n4 / lanes;                 // full pipelined stages

    ghmc_zero<<<1, 32, 0, stream>>>(g_cnt, g_sum);
    ghmc_pass1<<<BLOCKS, TPB, 0, stream>>>(pred, target, g_cnt, g_sum,
                                           n4, S, shiftC, maskC);
    ghmc_finalize<<<1, 32, 0, stream>>>(g_cnt, g_sum, (float*)d_out);
}